// Recommender_40475771797738
// MI455X (gfx1250) — compile-verified
//
#include <hip/hip_runtime.h>
#include <hip/hip_bf16.h>

// Problem dims (fixed by the reference)
#define QROWS 256
#define KDIM  512
#define NTRAIN 150000
#define NPAD  262144          // next pow2 >= NTRAIN (2^18)
#define EPSV  1e-8f

typedef __attribute__((ext_vector_type(2))) float v2f;
typedef __attribute__((ext_vector_type(8))) float v8f;
typedef unsigned long long u64;
typedef unsigned int u32;

#if __has_builtin(__builtin_amdgcn_global_load_async_to_lds_b128) && \
    __has_builtin(__builtin_amdgcn_s_wait_asynccnt)
#define HAS_ASYNC_LDS 1
// Parameter types per hipcc diagnostic: v4i addrspace(1)* src, v4i addrspace(3)* dst
typedef int v4i __attribute__((vector_size(16)));
typedef __attribute__((address_space(1))) v4i* gas_v4i;
typedef __attribute__((address_space(3))) v4i* las_v4i;
#else
#define HAS_ASYNC_LDS 0
#endif

// ---------------------------------------------------------------------------
// Key packing: ascending u64 sort of packed == descending score, stable index
// ---------------------------------------------------------------------------
__device__ __forceinline__ u64 pack_desc(float s, u32 idx) {
    u32 u   = __float_as_uint(s);
    u32 asc = u ^ ((u >> 31) ? 0xFFFFFFFFu : 0x80000000u);  // float -> asc uint
    u32 key = ~asc;                                          // invert -> descending
    return ((u64)key << 32) | idx;
}

__device__ __forceinline__ float unpack_val(u64 p) {
    u32 asc = ~(u32)(p >> 32);
    u32 u   = (asc >> 31) ? (asc ^ 0x80000000u) : ~asc;
    return __uint_as_float(u);
}

// ---------------------------------------------------------------------------
// Row L2 norms: one wave32 per row, float4 loads + shfl_xor butterfly
// ---------------------------------------------------------------------------
__global__ void rec_row_norms(const float* __restrict__ src,
                              float* __restrict__ dst, int rows) {
    int wavesPerBlock = blockDim.x >> 5;
    int row  = blockIdx.x * wavesPerBlock + (threadIdx.x >> 5);
    int lane = threadIdx.x & 31;
    if (row >= rows) return;
    const float4* r4 = (const float4*)(src + (size_t)row * KDIM);
    float acc = 0.f;
    #pragma unroll
    for (int i = lane; i < KDIM / 4; i += 32) {
        float4 v = r4[i];
        acc += v.x * v.x + v.y * v.y + v.z * v.z + v.w * v.w;
    }
    #pragma unroll
    for (int m = 16; m >= 1; m >>= 1) acc += __shfl_xor(acc, m, 32);
    if (lane == 0) dst[row] = sqrtf(acc);
}

// ---------------------------------------------------------------------------
// Fill sort padding [NTRAIN, NPAD) with +inf sentinels (sort to the end)
// ---------------------------------------------------------------------------
__global__ void rec_fill_pad(u64* __restrict__ packed) {
    const int padN = NPAD - NTRAIN;
    size_t t = (size_t)blockIdx.x * blockDim.x + threadIdx.x;
    size_t total = (size_t)QROWS * padN;
    if (t >= total) return;
    size_t row = t / padN, col = NTRAIN + t % padN;
    packed[row * NPAD + col] = 0xFFFFFFFFFFFFFFFFull;
}

// ---------------------------------------------------------------------------
// GEMM (fp32 WMMA 16x16x4) + cosine scaling + key packing
// Block: 256 threads = 8 waves. Each wave owns one 16(q) x 16(n) C tile.
// Grid: x = q-tile (16, fastest) so consecutive blocks share the same train
//       slab in L2; y = group of 8 n-tiles.
// Query tile (16x512) staged once in padded LDS, reused by all 8 waves.
// ---------------------------------------------------------------------------
#define LDSQ_STRIDE (KDIM + 4)   // +4 floats: bank-conflict-free A reads

__global__ void rec_gemm_pack(const float* __restrict__ query,
                              const float* __restrict__ train,
                              const float* __restrict__ qn,
                              const float* __restrict__ tn,
                              u64* __restrict__ packed) {
    __shared__ float ldsQ[16 * LDSQ_STRIDE];
    __shared__ float ldsQN[16];

    const int qbase = blockIdx.x * 16;

    // Stage 16x512 query tile into LDS (coalesced global, linear LDS)
    for (int e = threadIdx.x; e < 16 * KDIM; e += blockDim.x) {
        int r = e >> 9, c = e & (KDIM - 1);
        ldsQ[r * LDSQ_STRIDE + c] = query[(size_t)(qbase + r) * KDIM + c];
    }
    if (threadIdx.x < 16) ldsQN[threadIdx.x] = qn[qbase + threadIdx.x];
    __syncthreads();

    const int wave = threadIdx.x >> 5;
    const int lane = threadIdx.x & 31;
    const int half = lane >> 4;         // 0: lanes 0-15, 1: lanes 16-31
    const int l16  = lane & 15;
    const int koff = half * 2;          // A/B fragment K offset per ISA layout

    const int ntile = (blockIdx.y * 8 + wave) * 16;
    if (ntile >= NTRAIN) return;        // wave-uniform exit (after barrier)

    const int  n     = ntile + l16;
    const bool valid = (n < NTRAIN);
    const float* trow = train + (size_t)(valid ? n : 0) * KDIM;
    const float* arow = &ldsQ[l16 * LDSQ_STRIDE];

    v8f c = {};
    #pragma unroll 8
    for (int kk = 0; kk < KDIM; kk += 4) {
        if ((kk & 63) == 0)
            __builtin_prefetch(trow + kk + 128, 0, 3);   // WGP-scope prefetch
        v2f a, b;
        a.x = arow[kk + koff];           // A 16x4 f32: M=l16, K=koff(+1)
        a.y = arow[kk + koff + 1];
        b.x = trow[kk + koff];           // B 4x16 f32: N=l16, K=koff(+1)
        b.y = trow[kk + koff + 1];
        // D = A x B + C   (V_WMMA_F32_16X16X4_F32)
        c = __builtin_amdgcn_wmma_f32_16x16x4_f32(
                /*neg_a=*/false, a, /*neg_b=*/false, b,
                /*c_mod=*/(short)0, c, /*reuse_a=*/false, /*reuse_b=*/false);
    }

    const float tnv = tn[valid ? n : 0];
    if (valid) {
        #pragma unroll
        for (int v = 0; v < 8; v++) {
            int row = v + half * 8;                  // C layout: M=v(+8), N=l16
            float denom = fmaxf(ldsQN[row] * tnv, EPSV);
            float s = c[v] / denom;
            packed[(size_t)(qbase + row) * NPAD + n] = pack_desc(s, (u32)n);
        }
    }
}

// ---------------------------------------------------------------------------
// Bitonic sort, LDS-local phases: 2048-element blocks, 1024 threads.
// Blocks are staged global->LDS via GLOBAL_LOAD_ASYNC_TO_LDS_B128 (ASYNCcnt)
// when the builtin exists; otherwise through VGPRs.
// ---------------------------------------------------------------------------
__device__ __forceinline__ void cmpxchg(u64& x, u64& y, bool asc) {
    bool sw = asc ? (x > y) : (x < y);
    if (sw) { u64 t = x; x = y; y = t; }
}

__device__ __forceinline__ void stage_block_in(u64* sh, u64* p, int t) {
#if HAS_ASYNC_LDS
    // one b128 async copy per thread covers its two u64 elements
    __builtin_amdgcn_global_load_async_to_lds_b128(
        (gas_v4i)(p + 2 * t), (las_v4i)(sh + 2 * t),
        /*offset=*/0, /*cpol=*/0);
    __builtin_amdgcn_s_wait_asynccnt(0);
#else
    sh[2 * t] = p[2 * t];
    sh[2 * t + 1] = p[2 * t + 1];
#endif
}

__global__ __launch_bounds__(1024)
void rec_local_sort(u64* __restrict__ data) {
    __shared__ u64 sh[2048];
    const int t = threadIdx.x;
    const unsigned base = blockIdx.x * 2048u;
    u64* p = data + (size_t)blockIdx.y * NPAD + base;
    stage_block_in(sh, p, t);
    __syncthreads();
    for (unsigned k = 2; k <= 2048; k <<= 1) {
        for (unsigned j = k >> 1; j >= 1; j >>= 1) {
            unsigned i = ((t & ~(j - 1)) << 1) | (t & (j - 1));
            bool asc = (((base + i) & k) == 0);
            u64 x = sh[i], y = sh[i + j];
            cmpxchg(x, y, asc);
            sh[i] = x; sh[i + j] = y;
            __syncthreads();
        }
    }
    p[t] = sh[t];  p[t + 1024] = sh[t + 1024];
}

__global__ __launch_bounds__(1024)
void rec_local_merge(u64* __restrict__ data, unsigned k) {
    __shared__ u64 sh[2048];
    const int t = threadIdx.x;
    const unsigned base = blockIdx.x * 2048u;
    u64* p = data + (size_t)blockIdx.y * NPAD + base;
    stage_block_in(sh, p, t);
    __syncthreads();
    const bool asc = ((base & k) == 0);   // k >= 4096 -> uniform per block
    for (unsigned j = 1024; j >= 1; j >>= 1) {
        unsigned i = ((t & ~(j - 1)) << 1) | (t & (j - 1));
        u64 x = sh[i], y = sh[i + j];
        cmpxchg(x, y, asc);
        sh[i] = x; sh[i + j] = y;
        __syncthreads();
    }
    p[t] = sh[t];  p[t + 1024] = sh[t + 1024];
}

__global__ void rec_global_step(u64* __restrict__ data, unsigned k, unsigned j) {
    unsigned pair = blockIdx.x * blockDim.x + threadIdx.x;   // NPAD/2 pairs/row
    u64* row = data + (size_t)blockIdx.y * NPAD;
    unsigned i = ((pair & ~(j - 1)) << 1) | (pair & (j - 1));
    bool asc = ((i & k) == 0);
    u64 x = row[i], y = row[i | j];
    cmpxchg(x, y, asc);
    row[i] = x; row[i | j] = y;
}

// ---------------------------------------------------------------------------
// Unpack first NTRAIN (ascending u64 == descending score) into (vals, idx)
// ---------------------------------------------------------------------------
__global__ void rec_unpack(const u64* __restrict__ packed,
                           float* __restrict__ vals, int* __restrict__ idx) {
    size_t t = (size_t)blockIdx.x * blockDim.x + threadIdx.x;
    size_t total = (size_t)QROWS * NTRAIN;
    if (t >= total) return;
    size_t row = t / NTRAIN, r = t % NTRAIN;
    u64 p = packed[row * NPAD + r];
    vals[t] = unpack_val(p);
    idx[t]  = (int)(u32)p;
}

// ---------------------------------------------------------------------------
// Host launcher. Needs ws_size >= ~537.5 MB:
//   [0, 600000)            train norms
//   [600064, 601088)       query norms
//   [602112, 602112+512MB) packed key|index buffer (256 x 262144 x 8B)
// ---------------------------------------------------------------------------
extern "C" void kernel_launch(void* const* d_in, const int* in_sizes, int n_in,
                              void* d_out, int out_size, void* d_ws, size_t ws_size,
                              hipStream_t stream) {
    const float* query = (const float*)d_in[0];
    const float* train = (const float*)d_in[1];

    char*  base   = (char*)d_ws;
    float* tnorm  = (float*)base;                 // 150000 floats
    float* qnorm  = (float*)(base + 600064);      // 256 floats
    u64*   packed = (u64*)  (base + 602112);      // 256 * 262144 u64

    float* vals = (float*)d_out;
    int*   idx  = (int*)((float*)d_out + (size_t)QROWS * NTRAIN);

    // 1) Row norms (wave32 per row)
    rec_row_norms<<<(NTRAIN + 7) / 8, 256, 0, stream>>>(train, tnorm, NTRAIN);
    rec_row_norms<<<(QROWS + 7) / 8, 256, 0, stream>>>(query, qnorm, QROWS);

    // 2) Sentinel-fill sort padding
    {
        size_t total = (size_t)QROWS * (NPAD - NTRAIN);
        rec_fill_pad<<<(unsigned)((total + 255) / 256), 256, 0, stream>>>(packed);
    }

    // 3) fp32 WMMA GEMM + cosine scale + key packing
    {
        dim3 grid(QROWS / 16 /*=16, fastest: L2 reuse of train*/,
                  (9375 + 7) / 8 /*=1172 groups of 8 n-tiles*/);
        rec_gemm_pack<<<grid, 256, 0, stream>>>(query, train, qnorm, tnorm, packed);
    }

    // 4) Bitonic sort: local (k<=2048) then global+local-merge phases
    rec_local_sort<<<dim3(NPAD / 2048, QROWS), 1024, 0, stream>>>(packed);
    for (unsigned k = 4096; k <= (unsigned)NPAD; k <<= 1) {
        for (unsigned j = k >> 1; j >= 2048; j >>= 1)
            rec_global_step<<<dim3(NPAD / 2 / 256, QROWS), 256, 0, stream>>>(packed, k, j);
        rec_local_merge<<<dim3(NPAD / 2048, QROWS), 1024, 0, stream>>>(packed, k);
    }

    // 5) Unpack top-NTRAIN per row into (vals, idx)
    {
        size_t total = (size_t)QROWS * NTRAIN;
        rec_unpack<<<(unsigned)((total + 255) / 256), 256, 0, stream>>>(packed, vals, idx);
    }
}